// FixedProjectionBlock_44976897524206
// MI455X (gfx1250) — compile-verified
//
#include <hip/hip_runtime.h>
#include <hip/hip_bf16.h>
#include <math.h>

#define PI_F     3.14159265358979323846f
#define TWOPI_F  6.28318530717958647692f

typedef float v2f __attribute__((ext_vector_type(2)));
typedef float v8f __attribute__((ext_vector_type(8)));

#define B_DIM 2
#define L_DIM 1024
#define D_DIM 512
#define TP_DIM 16
#define NS_DIM 4
#define PP_DIM 4
#define P_DIM 16
#define PH_STRIDE 80   // 16 kpc +16 kps +16 qpc +16 qps +4 jkc +4 jks +4 jqc +4 jqs
#define POS_STRIDE 32  // 16 pfc + 16 pfs

__device__ __forceinline__ float sigmoidf(float v) { return 1.0f / (1.0f + expf(-v)); }

// ---------------------------------------------------------------------------
// fp32 GEMM via V_WMMA_F32_16X16X4_F32 with LDS-staged A panel.
// Block covers ONE 16-row A panel; waves/block == N/(16*NT), wave w computes
// the 16 x (16*NT) tile at columns w*16*NT.  A panel (16 x K) is staged into
// LDS once (coalesced), padded row stride K+2 for bank-conflict-free reads.
// op==1 applies tanh(.)*pi epilogue.  bias/resid nullable.
// grid.x = M/16, block = (N/(16*NT))*32 threads, dyn-LDS = 16*(K+2)*4 bytes.
// ---------------------------------------------------------------------------
template <int NT>
__global__ void gemm_f32_wmma(const float* __restrict__ A,
                              const float* __restrict__ Bm,
                              const float* __restrict__ bias,
                              const float* __restrict__ resid,
                              float* __restrict__ C,
                              int M, int N, int K, int op)
{
    extern __shared__ float alds[];           // [16][K+2]
    const int wave = threadIdx.x >> 5;
    const int lane = threadIdx.x & 31;
    const int tm   = blockIdx.x << 4;
    const int tn0  = wave * (16 * NT);
    const int hi   = lane >> 4;               // 0: K pair {0,1}, 1: K pair {2,3}
    const int row  = lane & 15;
    const int ldk  = K + 2;

    // cooperative, coalesced stage of the A panel (16*K/2 float2 elements)
    const int nf2 = (16 * K) >> 1;
    for (int idx = threadIdx.x; idx < nf2; idx += blockDim.x) {
        const int r  = idx / (K >> 1);
        const int c2 = (idx - r * (K >> 1)) << 1;
        const float2 v = *(const float2*)(A + (size_t)(tm + r) * K + c2);
        alds[r * ldk + c2]     = v.x;
        alds[r * ldk + c2 + 1] = v.y;
    }
    __syncthreads();

    v8f acc[NT];
#pragma unroll
    for (int t = 0; t < NT; ++t)
        acc[t] = (v8f){0.f, 0.f, 0.f, 0.f, 0.f, 0.f, 0.f, 0.f};

    const float* Arow = alds + row * ldk + (hi << 1);
    const float* Bcol = Bm + (size_t)(hi << 1) * N + (tn0 + row);
    __builtin_prefetch(Bcol, 0, 1);           // global_prefetch_b8: warm B panel

    for (int kb = 0; kb < K; kb += 4) {
        const float2 av = *(const float2*)(Arow + kb);   // LDS, bank-spread
        v2f a; a.x = av.x; a.y = av.y;
        const float* bp = Bcol + (size_t)kb * N;
#pragma unroll
        for (int t = 0; t < NT; ++t) {
            v2f b;
            b.x = bp[t * 16];
            b.y = bp[t * 16 + N];
            acc[t] = __builtin_amdgcn_wmma_f32_16x16x4_f32(false, a, false, b,
                                                           (short)0, acc[t],
                                                           false, false);
        }
    }

#pragma unroll
    for (int t = 0; t < NT; ++t) {
#pragma unroll
        for (int r = 0; r < 8; ++r) {
            const int m = tm + r + (hi << 3);
            const int n = tn0 + t * 16 + row;
            float val = acc[t][r];
            if (bias)     val += bias[n];
            if (op == 1)  val  = tanhf(val) * PI_F;
            if (resid)    val += resid[(size_t)m * N + n];
            C[(size_t)m * N + n] = val;
        }
    }
}

// ---------------------------------------------------------------------------
// Phasor table prep: PH[b*L+l][80] = {kpc[16],kps[16],qpc[16],qps[16],
//                                     jkc[4],jks[4],jqc[4],jqs[4]}
// ---------------------------------------------------------------------------
__global__ void phasor_prep_kernel(const float* __restrict__ TK,
                                   const float* __restrict__ TQ,
                                   float* __restrict__ PH)
{
    const int idx = blockIdx.x * blockDim.x + threadIdx.x;
    if (idx >= B_DIM * L_DIM) return;
    const float* tk = TK + (size_t)idx * TP_DIM;
    const float* tq = TQ + (size_t)idx * TP_DIM;
    float* ph = PH + (size_t)idx * PH_STRIDE;
#pragma unroll
    for (int i = 0; i < 16; ++i) {
        float s, c;
        sincosf(tk[i], &s, &c); ph[i]      = c; ph[16 + i] = s;
        sincosf(tq[i], &s, &c); ph[32 + i] = c; ph[48 + i] = s;
    }
#pragma unroll
    for (int pp = 0; pp < 4; ++pp) {
        float s, c;
        sincosf(tk[pp] + tk[4 + pp] + tk[8 + pp] + tk[12 + pp], &s, &c);
        ph[64 + pp] = c; ph[68 + pp] = s;
        sincosf(tq[pp] + tq[4 + pp] + tq[8 + pp] + tq[12 + pp], &s, &c);
        ph[72 + pp] = c; ph[76 + pp] = s;
    }
}

// positional phasors: POS[l][32] = {pfc[16], pfs[16]}
__global__ void pos_prep_kernel(const float* __restrict__ pos_freqs,
                                float* __restrict__ POS)
{
    const int idx = blockIdx.x * blockDim.x + threadIdx.x;
    if (idx >= L_DIM * P_DIM) return;
    const int l = idx >> 4;
    const int p = idx & 15;
    float s, c;
    sincosf((float)l * pos_freqs[p] * TWOPI_F, &s, &c);
    POS[(size_t)l * POS_STRIDE + p]      = c;
    POS[(size_t)l * POS_STRIDE + 16 + p] = s;
}

// ---------------------------------------------------------------------------
// Gate MLP head: gate[row] = sigmoid( gelu(G1[row,:]) @ g2_w + g2_b )
// ---------------------------------------------------------------------------
__global__ void gate_kernel(const float* __restrict__ G1,
                            const float* __restrict__ g2w,
                            const float* __restrict__ g2b,
                            float* __restrict__ gate, int H)
{
    __shared__ float red[256];
    const int rowi = blockIdx.x;
    const int tid  = threadIdx.x;
    float t  = G1[(size_t)rowi * H + tid];
    float ge = 0.5f * t * (1.0f + erff(t * 0.7071067811865475f));  // exact gelu
    red[tid] = ge * g2w[tid];
    __syncthreads();
    for (int s = 128; s > 0; s >>= 1) {
        if (tid < s) red[tid] += red[tid + s];
        __syncthreads();
    }
    if (tid == 0) gate[rowi] = sigmoidf(red[0] + g2b[0]);
}

// ---------------------------------------------------------------------------
// Surprise write-gate scan.  |km*conj(jq)| == |km| (unit phasor), so only the
// shifted cumsum of the joint key phasor matters.  One wave per batch.
// ---------------------------------------------------------------------------
__global__ void wg_scan_kernel(const float* __restrict__ PH,
                               const float* __restrict__ sscale,
                               const float* __restrict__ sbias,
                               const float* __restrict__ rscale,
                               const float* __restrict__ rthr,
                               float* __restrict__ wg)
{
    const int b    = blockIdx.x;
    const int lane = threadIdx.x;
    const float ss = sscale[0], sb = sbias[0];
    const float sc = fminf(fmaxf(rscale[0], 1.0f), 20.0f);
    const float th = fminf(fmaxf(rthr[0], 0.1f), 0.9f);
    float kmr = 0.f, kmi = 0.f;
    for (int l = 0; l < L_DIM; ++l) {
        float m = (lane < PP_DIM) ? sqrtf(kmr * kmr + kmi * kmi) : 0.f;
        m += __shfl_xor(m, 1);
        m += __shfl_xor(m, 2);
        if (lane == 0) {
            float mag  = m * 0.25f;
            float posn = fmaxf((float)l, 1.0f);
            float nres = mag * rsqrtf(posn);
            float sur  = 0.5f * (1.0f - tanhf(sc * (nres - th)));
            wg[b * L_DIM + l] = sigmoidf(ss * (sur - 0.5f) + sb);
        }
        if (lane < PP_DIM) {
            const float* ph = PH + (size_t)(b * L_DIM + l) * PH_STRIDE;
            kmr += ph[64 + lane];   // jkc
            kmi += ph[68 + lane];   // jks
        }
    }
}

// ---------------------------------------------------------------------------
// Main holographic scan.  One thread per channel d; 36 complex accumulators in
// registers.  All per-step phasors come from the precomputed tables via
// block-uniform loads -> no barriers, no trig in the sequential loop.
// grid = (D/64, B), block = 64
// ---------------------------------------------------------------------------
__global__ void holo_scan_kernel(const float* __restrict__ PH,
                                 const float* __restrict__ POS,
                                 const float* __restrict__ V,
                                 const float* __restrict__ wg,
                                 const float* __restrict__ gate,
                                 const float* __restrict__ set_weights,
                                 const float* __restrict__ pos_weight,
                                 float* __restrict__ y)
{
    const int b = blockIdx.y;
    const int d = blockIdx.x * blockDim.x + threadIdx.x;

    // softmax(set_weights)
    float w0 = set_weights[0], w1 = set_weights[1],
          w2 = set_weights[2], w3 = set_weights[3];
    float mx = fmaxf(fmaxf(w0, w1), fmaxf(w2, w3));
    float e0 = expf(w0 - mx), e1 = expf(w1 - mx),
          e2 = expf(w2 - mx), e3 = expf(w3 - mx);
    float inv = 1.0f / (e0 + e1 + e2 + e3);
    float w[4] = {e0 * inv, e1 * inv, e2 * inv, e3 * inv};
    const float posw = sigmoidf(pos_weight[0]);

    float cmr[4]  = {0.f, 0.f, 0.f, 0.f};
    float cmi[4]  = {0.f, 0.f, 0.f, 0.f};
    float pmr[16] = {0.f}; float pmi[16] = {0.f};
    float ppr[16] = {0.f}; float ppi[16] = {0.f};

    for (int l = 0; l < L_DIM; ++l) {
        const size_t rl = (size_t)(b * L_DIM + l);
        const float* ph = PH  + rl * PH_STRIDE;
        const float* ps = POS + (size_t)l * POS_STRIDE;

        const float v  = V[rl * D_DIM + d];
        const float vg = v * wg[rl];

        // cross (joint) path: Re(mem * conj(jq))
        float cross = 0.f;
#pragma unroll
        for (int pp = 0; pp < 4; ++pp) {
            cmr[pp] += ph[64 + pp] * vg;
            cmi[pp] += ph[68 + pp] * vg;
            cross   += cmr[pp] * ph[72 + pp] + cmi[pp] * ph[76 + pp];
        }
        float tot = cross;

        // per-bank path
#pragma unroll
        for (int ns = 0; ns < 4; ++ns) {
            float bank = 0.f;
#pragma unroll
            for (int pp = 0; pp < 4; ++pp) {
                int i = ns * 4 + pp;
                pmr[i] += ph[i]      * vg;
                pmi[i] += ph[16 + i] * vg;
                bank   += pmr[i] * ph[32 + i] + pmi[i] * ph[48 + i];
            }
            tot += w[ns] * bank;
        }
        tot *= 0.2f;  // 1/(NS+1)

        // positional path (ungated V)
        float ret = 0.f;
#pragma unroll
        for (int p = 0; p < 16; ++p) {
            ppr[p] += ps[p]      * v;
            ppi[p] += ps[16 + p] * v;
            ret    += ppr[p] * ps[p] + ppi[p] * ps[16 + p];
        }

        const float g   = gate[rl];
        const float out = g * tot + (1.0f - g) * (posw * ret);
        y[rl * D_DIM + d] = out * rsqrtf((float)(l + 1) * (float)PP_DIM);
    }
}

// ---------------------------------------------------------------------------
// LayerNorm over D=512, block=256 (2 elements/thread), one row per block
// ---------------------------------------------------------------------------
__global__ void ln_kernel(const float* __restrict__ y,
                          const float* __restrict__ lng,
                          const float* __restrict__ lnb,
                          float* __restrict__ yn)
{
    __shared__ float s1[256], s2[256];
    __shared__ float smu, sinv;
    const int rowi = blockIdx.x;
    const int tid  = threadIdx.x;
    const size_t base = (size_t)rowi * D_DIM;
    float a = y[base + tid];
    float c = y[base + tid + 256];
    s1[tid] = a + c;
    s2[tid] = a * a + c * c;
    __syncthreads();
    for (int s = 128; s > 0; s >>= 1) {
        if (tid < s) { s1[tid] += s1[tid + s]; s2[tid] += s2[tid + s]; }
        __syncthreads();
    }
    if (tid == 0) {
        float mu  = s1[0] * (1.0f / D_DIM);
        float var = s2[0] * (1.0f / D_DIM) - mu * mu;
        smu  = mu;
        sinv = rsqrtf(var + 1e-5f);
    }
    __syncthreads();
    const float mu = smu, iv = sinv;
    yn[base + tid]       = (a - mu) * iv * lng[tid]       + lnb[tid];
    yn[base + tid + 256] = (c - mu) * iv * lng[tid + 256] + lnb[tid + 256];
}

// ---------------------------------------------------------------------------
extern "C" void kernel_launch(void* const* d_in, const int* in_sizes, int n_in,
                              void* d_out, int out_size, void* d_ws, size_t ws_size,
                              hipStream_t stream)
{
    (void)in_sizes; (void)n_in; (void)out_size; (void)ws_size;

    const float* x        = (const float*)d_in[0];
    const float* key_proj = (const float*)d_in[1];
    const float* qry_proj = (const float*)d_in[2];
    // d_in[3] = ltm_proj (dead: ltm_ret == 0)
    const float* v_w      = (const float*)d_in[4];
    const float* v_b      = (const float*)d_in[5];
    const float* ln_g     = (const float*)d_in[6];
    const float* ln_b     = (const float*)d_in[7];
    const float* out_w    = (const float*)d_in[8];
    const float* out_b    = (const float*)d_in[9];
    const float* set_w    = (const float*)d_in[10];
    const float* pfreq    = (const float*)d_in[11];
    const float* pos_wt   = (const float*)d_in[12];
    const float* g1_w     = (const float*)d_in[13];
    const float* g1_b     = (const float*)d_in[14];
    const float* g2_w     = (const float*)d_in[15];
    const float* g2_b     = (const float*)d_in[16];
    // d_in[17..18] ltmv (dead)
    const float* s_scale  = (const float*)d_in[19];
    const float* s_bias   = (const float*)d_in[20];
    const float* r_scale  = (const float*)d_in[21];
    const float* r_thr    = (const float*)d_in[22];
    // d_in[23..27] LTM params (dead)

    const int M = B_DIM * L_DIM;   // 2048
    const int D = D_DIM;           // 512
    const int K = D_DIM;           // 512 (all GEMMs here have K=512)
    const size_t ldsA = (size_t)16 * (K + 2) * sizeof(float);  // 32,896 B

    float* ws   = (float*)d_ws;
    float* TK   = ws;                            // 2048*16
    float* TQ   = TK   + (size_t)M * TP_DIM;     // 2048*16
    float* V    = TQ   + (size_t)M * TP_DIM;     // 2048*512
    float* G1   = V    + (size_t)M * D;          // 2048*256
    float* gate = G1   + (size_t)M * 256;        // 2048
    float* wg   = gate + M;                      // 2048
    float* y    = wg   + M;                      // 2048*512
    float* yn   = y    + (size_t)M * D;          // 2048*512
    float* PH   = yn   + (size_t)M * D;          // 2048*80
    float* POS  = PH   + (size_t)M * PH_STRIDE;  // 1024*32
    float* out  = (float*)d_out;

    // grid.x = M/16 panels; block = (N/(16*NT)) waves * 32
    // phase projections (N=16, NT=1 -> 1 wave/block): tanh(.)*pi epilogue
    gemm_f32_wmma<1><<<M / 16, 32, ldsA, stream>>>(
        x, key_proj, nullptr, nullptr, TK, M, TP_DIM, K, 1);
    gemm_f32_wmma<1><<<M / 16, 32, ldsA, stream>>>(
        x, qry_proj, nullptr, nullptr, TQ, M, TP_DIM, K, 1);
    // V = x @ v_w + v_b   (N=512, NT=4 -> 8 waves/block)
    gemm_f32_wmma<4><<<M / 16, 256, ldsA, stream>>>(
        x, v_w, v_b, nullptr, V, M, D, K, 0);
    // G1 = x @ g1_w + g1_b (N=256, NT=4 -> 4 waves/block)
    gemm_f32_wmma<4><<<M / 16, 128, ldsA, stream>>>(
        x, g1_w, g1_b, nullptr, G1, M, 256, K, 0);

    // phasor tables
    phasor_prep_kernel<<<(M + 255) / 256, 256, 0, stream>>>(TK, TQ, PH);
    pos_prep_kernel<<<(L_DIM * P_DIM + 255) / 256, 256, 0, stream>>>(pfreq, POS);

    // gate head
    gate_kernel<<<M, 256, 0, stream>>>(G1, g2_w, g2_b, gate, 256);
    // surprise gate scan (scalar per (b,l))
    wg_scan_kernel<<<B_DIM, 32, 0, stream>>>(PH, s_scale, s_bias, r_scale, r_thr, wg);
    // main holographic scan (barrier-free)
    holo_scan_kernel<<<dim3(D / 64, B_DIM), 64, 0, stream>>>(
        PH, POS, V, wg, gate, set_w, pos_wt, y);
    // LayerNorm
    ln_kernel<<<M, 256, 0, stream>>>(y, ln_g, ln_b, yn);
    // out = x + yn @ out_w + out_b
    gemm_f32_wmma<4><<<M / 16, 256, ldsA, stream>>>(
        yn, out_w, out_b, x, out, M, D, K, 0);
}